// MultiSequenceRotaryPositionalEncoding_72138270703623
// MI455X (gfx1250) — compile-verified
//
#include <hip/hip_runtime.h>
#include <math.h>

// Problem constants from the reference (L, N, H fixed; B derived at launch).
#define RL   8192          // sequence length L
#define RNH  1024          // N*H floats per (b,l) row
#define K1T  256           // threads for position kernel
#define CHNK (RL / K1T)    // 32 lengths per thread

typedef float v4f __attribute__((ext_vector_type(4)));

// ---------------------------------------------------------------------------
// Kernel 1: segment-reset positions.  One block per batch row.
// csum = inclusive cumsum(lengths); pos[j] = j - csum[seg-1] for the segment
// containing j (searchsorted right), 0 for j >= csum[L-1].
// ---------------------------------------------------------------------------
__global__ __launch_bounds__(K1T)
void rope_positions_kernel(const int* __restrict__ idx, float* __restrict__ posOut) {
  __shared__ int csum[RL];     // 32 KB
  __shared__ int part[K1T];

  const int b = blockIdx.x;
  const int t = threadIdx.x;
  const int* row = idx + (size_t)b * RL;

  // per-thread partial sums of a contiguous chunk of 32 lengths
  int s = 0;
  for (int i = 0; i < CHNK; ++i) s += row[t * CHNK + i];
  part[t] = s;
  __syncthreads();

  // serial exclusive scan of 256 partials (perf-irrelevant tiny kernel)
  if (t == 0) {
    int run = 0;
    for (int i = 0; i < K1T; ++i) { int v = part[i]; part[i] = run; run += v; }
  }
  __syncthreads();

  // inclusive cumsum into LDS
  int run = part[t];
  for (int i = 0; i < CHNK; ++i) {
    run += row[t * CHNK + i];
    csum[t * CHNK + i] = run;
  }
  __syncthreads();

  const int total = csum[RL - 1];

  // binary search per token (strided for coalesced pos stores)
  for (int i = 0; i < CHNK; ++i) {
    const int j = t + i * K1T;
    int lo = 0, hi = RL;                     // first idx with csum[idx] > j
    while (lo < hi) {
      int mid = (lo + hi) >> 1;
      if (csum[mid] <= j) lo = mid + 1; else hi = mid;
    }
    int seg   = lo < (RL - 1) ? lo : (RL - 1);
    int start = (seg > 0) ? csum[seg - 1] : 0;
    int p     = (j < total) ? (j - start) : 0;
    posOut[(size_t)b * RL + j] = (float)p;
  }
}

// ---------------------------------------------------------------------------
// Kernel 2: apply rotary embedding.  256 threads handle 2 (b,l) rows.
// Per row: 128 threads, thread u -> head n=u/8, quad q=u%8.
//   A = x[row, n, 4q..4q+3]        (lower half)
//   B = x[row, n, 32+4q..32+4q+3]  (upper half)
//   out_lo = A*c - B*s ; out_hi = B*c + A*s   with c,s = cos/sin(pos*invf[k])
// x quads are staged through LDS with CDNA5 async global->LDS b128 loads;
// the lead wave of each row computes the 32 (cos,sin) pairs meanwhile.
// ---------------------------------------------------------------------------
__global__ __launch_bounds__(256)
void rope_apply_kernel(const float* __restrict__ x,
                       const float* __restrict__ pos,
                       float* __restrict__ out) {
  __shared__ __align__(16) float sX[2][RNH];   // 8 KB staged x
  __shared__ __align__(16) float sC[2][32];
  __shared__ __align__(16) float sS[2][32];

  const int tid   = threadIdx.x;
  const int slot  = tid >> 7;                  // 0/1: which row of this block
  const int u     = tid & 127;
  const int n     = u >> 3;                    // head 0..15
  const int q     = u & 7;                     // quad 0..7 within lower half
  const int row   = blockIdx.x * 2 + slot;
  const size_t rowBase = (size_t)row * RNH;
  const int inRow = n * 64 + q * 4;            // float offset within row

  const float* gA = x + rowBase + inRow;       // lower-half quad
  const float* gB = gA + 32;                   // matching upper-half quad

  // Workgroup-relative LDS byte offsets (flat ptr low 32 bits == LDS offset,
  // ISA 10.2 aperture mapping).
  unsigned ldsA = (unsigned)(unsigned long long)(const void*)&sX[slot][inRow];
  unsigned ldsB = ldsA + 32u * 4u;

  // CDNA5 async copies: per-lane 16B global -> LDS, tracked by ASYNCcnt.
  asm volatile("global_load_async_to_lds_b128 %0, %1, off"
               :: "v"(ldsA), "v"(gA) : "memory");
  asm volatile("global_load_async_to_lds_b128 %0, %1, off"
               :: "v"(ldsB), "v"(gB) : "memory");

  // Overlap: lead wave of each row computes cos/sin for k = 0..31.
  if (u < 32) {
    float p = pos[row];
    // inv_freq[k] = 10000^(-k/32); double exp2 keeps angle error ~<1e-4 rad
    // even at pos=511 (k=0 is exactly 1.0, so the largest angles are exact).
    float invf = (float)exp2(-(double)u * 0.41524101186092029); // log2(1e4)/32
    float ang = p * invf;
    float sv, cv;
    sincosf(ang, &sv, &cv);
    sC[slot][u] = cv;
    sS[slot][u] = sv;
  }

  // Drain this wave's async copies, then barrier so sC/sS (and all staged
  // data) are visible block-wide.
  asm volatile("s_wait_asynccnt 0" ::: "memory");
  __syncthreads();

  v4f A  = *(const v4f*)&sX[slot][inRow];
  v4f Bv = *(const v4f*)&sX[slot][inRow + 32];
  v4f c  = *(const v4f*)&sC[slot][q * 4];
  v4f s  = *(const v4f*)&sS[slot][q * 4];

  v4f o1 = A * c - Bv * s;    // lower half
  v4f o2 = Bv * c + A * s;    // upper half

  // Streaming output: non-temporal 128-bit stores (no reuse; spare the L2).
  __builtin_nontemporal_store(o1, (v4f*)(out + rowBase + inRow));
  __builtin_nontemporal_store(o2, (v4f*)(out + rowBase + inRow + 32));
}

// ---------------------------------------------------------------------------
extern "C" void kernel_launch(void* const* d_in, const int* in_sizes, int n_in,
                              void* d_out, int out_size, void* d_ws, size_t ws_size,
                              hipStream_t stream) {
  (void)n_in; (void)out_size; (void)ws_size;

  const float* x   = (const float*)d_in[0];   // [B, L, N, H] f32
  const int*   idx = (const int*)d_in[1];     // [B, L] int
  float*       out = (float*)d_out;           // [B, L, N, H] f32
  float*       posw = (float*)d_ws;           // [B*L] f32 scratch

  const int rows = in_sizes[1];               // B * L
  const int B    = rows / RL;                 // 4

  rope_positions_kernel<<<B, K1T, 0, stream>>>(idx, posw);
  rope_apply_kernel<<<rows / 2, 256, 0, stream>>>(x, posw, out);
}